// NeuralBellmanFordNetworkTemporal_54082228191901
// MI455X (gfx1250) — compile-verified
//
#include <hip/hip_runtime.h>
#include <math.h>

#define NN 50000
#define BB 2
#define DD 64
#define EE 800000
#define RR 32
#define TDM 64
#define TMAX 1000
#define EPSV 1e-6f

typedef float v2f __attribute__((ext_vector_type(2)));
typedef float v8f __attribute__((ext_vector_type(8)));

// order-preserving float <-> uint key for integer max/min atomics
__device__ __forceinline__ unsigned fkey(float f) {
    unsigned u = __float_as_uint(f);
    unsigned mask = (unsigned)(((int)u) >> 31) | 0x80000000u;
    return u ^ mask;
}
__device__ __forceinline__ float funkey(unsigned u) {
    unsigned bits = (u & 0x80000000u) ? (u ^ 0x80000000u) : ~u;
    return __uint_as_float(bits);
}

// ---------------- precompute kernels ----------------

// rel_table[r][b][d] = sum_j query[b][j] * W_rel[(r*DD+d)*DD + j] + b_rel[r*DD+d]
__global__ void k_rel(const float* __restrict__ query, const float* __restrict__ W_rel,
                      const float* __restrict__ b_rel, float* __restrict__ rel_table) {
    int id = blockIdx.x * blockDim.x + threadIdx.x;
    if (id >= RR * BB * DD) return;
    int r = id / (BB * DD);
    int b = (id / DD) % BB;
    int d = id % DD;
    const float* w = W_rel + (size_t)(r * DD + d) * DD;
    const float* q = query + b * DD;
    float acc = b_rel[r * DD + d];
    #pragma unroll 8
    for (int j = 0; j < DD; ++j) acc += q[j] * w[j];
    rel_table[(r * BB + b) * DD + d] = acc;
}

// ttable[t][d] = sum_j te(t)[j] * W_time[d*TDM + j] + b_time[d]
__global__ void k_time(const float* __restrict__ W_time, const float* __restrict__ b_time,
                       float* __restrict__ ttable) {
    int id = blockIdx.x * blockDim.x + threadIdx.x;
    if (id >= TMAX * DD) return;
    int t = id / DD;
    int d = id % DD;
    const float step = 0.29710775393471559f; // log(10000)/31
    const float* w = W_time + (size_t)d * TDM;
    float tf = (float)t;
    float acc = b_time[d];
    #pragma unroll 4
    for (int j = 0; j < 32; ++j) {
        float invf = __expf(-(float)j * step);
        float ang = tf * invf;
        acc += __sinf(ang) * w[j] + __cosf(ang) * w[j + 32];
    }
    ttable[t * DD + d] = acc;
}

// wtab[t][b] = (qt[b] >= t) ? 2^(-(qt[b]-t)/200) : 0
__global__ void k_w(const int* __restrict__ query_time, float* __restrict__ wtab) {
    int id = blockIdx.x * blockDim.x + threadIdx.x;
    if (id >= TMAX * BB) return;
    int t = id / BB;
    int b = id % BB;
    int dt = query_time[b] - t;
    wtab[id] = (dt >= 0) ? exp2f(-(float)dt * (1.0f / 200.0f)) : 0.0f;
}

// init stats with the boundary self-loop term (weight 1), zero counters
__global__ void k_init(const float* __restrict__ boundary, float* __restrict__ sums,
                       float* __restrict__ sqs, unsigned* __restrict__ mxk,
                       unsigned* __restrict__ mnk, float* __restrict__ cnt,
                       float* __restrict__ degw, float* __restrict__ scalesum) {
    int id = blockIdx.x * blockDim.x + threadIdx.x;
    if (id < NN * BB * DD) {
        float m = boundary[id];
        sums[id] = m;
        sqs[id] = m * m;
        unsigned k = fkey(m);
        mxk[id] = k;
        mnk[id] = k;
    }
    if (id < NN) { cnt[id] = 1.0f; degw[id] = 0.0f; }
    if (id == 0) scalesum[0] = 0.0f;
}

// ---------------- edge scatter: one wave per edge ----------------
__global__ void k_edge(const float* __restrict__ input, const float* __restrict__ edge_weight,
                       const int* __restrict__ node_in, const int* __restrict__ node_out,
                       const int* __restrict__ relation, const int* __restrict__ edge_time,
                       const float* __restrict__ rel_table, const float* __restrict__ ttable,
                       const float* __restrict__ wtab, float* __restrict__ sums,
                       float* __restrict__ sqs, unsigned* __restrict__ mxk,
                       unsigned* __restrict__ mnk, float* __restrict__ cnt,
                       float* __restrict__ degw) {
    int wid = (blockIdx.x * blockDim.x + threadIdx.x) >> 5;
    int lane = threadIdx.x & 31;
    if (wid >= EE) return;
    int nin  = node_in[wid];
    int nout = node_out[wid];
    int rel  = relation[wid];
    int t    = edge_time[wid];
    float ew = edge_weight[wid];
    const float* nip = input + (size_t)nin * BB * DD;
    const float* rp  = rel_table + (size_t)rel * BB * DD;
    const float* tt  = ttable + (size_t)t * DD;
    size_t obase = (size_t)nout * BB * DD;
    #pragma unroll
    for (int i = 0; i < 4; ++i) {
        int idx = i * 32 + lane;     // idx = b*64 + d
        int b = idx >> 6;
        int d = idx & 63;
        int d2 = d & 31;
        float nr = nip[b * DD + d2], ni = nip[b * DD + d2 + 32];
        float er = rp[b * DD + d2],  ei = rp[b * DD + d2 + 32];
        float m = (d < 32) ? (nr * er - ni * ei) : (nr * ei + ni * er);
        m += tt[d];
        m *= wtab[t * BB + b];
        float mw = m * ew;
        atomicAdd(&sums[obase + idx], mw);
        atomicAdd(&sqs[obase + idx], m * m * ew);
        unsigned k = fkey(mw);
        atomicMax(&mxk[obase + idx], k);
        atomicMin(&mnk[obase + idx], k);
    }
    if (lane == 0) {
        atomicAdd(&cnt[nout], 1.0f);
        atomicAdd(&degw[nout], ew);
    }
}

// scale_raw[n] = log(degw+1); accumulate sum for mean
__global__ void k_scale(const float* __restrict__ degw, float* __restrict__ scale_raw,
                        float* __restrict__ scalesum) {
    int n = blockIdx.x * blockDim.x + threadIdx.x;
    if (n >= NN) return;
    float s = logf(degw[n] + 1.0f);
    scale_raw[n] = s;
    atomicAdd(scalesum, s);
}

// ---------------- fused feature build + WMMA GEMM ----------------
// One wave per 16-row tile of (N*B) x 832 features; 4 column tiles of 16.
#define XST 836   // row stride in floats (even -> 8B aligned v2f loads)

__global__ __launch_bounds__(32)
void k_gemm(const float* __restrict__ input, const float* __restrict__ sums,
            const float* __restrict__ sqs, const unsigned* __restrict__ mxk,
            const unsigned* __restrict__ mnk, const float* __restrict__ cnt,
            const float* __restrict__ scale_raw, const float* __restrict__ scalesum,
            const float* __restrict__ W_lin, const float* __restrict__ b_lin,
            float* __restrict__ out) {
    __shared__ float Xs[16 * XST];
    __shared__ float sc[16][3];

    int tile = blockIdx.x;
    int lane = threadIdx.x;
    float inv_mean = (float)NN / scalesum[0];

    if (lane < 16) {
        int g = tile * 16 + lane;
        int n = g >> 1;               // BB == 2
        float s = scale_raw[n] * inv_mean;
        sc[lane][0] = 1.0f;
        sc[lane][1] = s;
        sc[lane][2] = 1.0f / fmaxf(s, 0.01f);
    }
    __syncthreads();

    // Build X[16][832] = [ input(64) | update(768) ] in LDS
    for (int ii = lane; ii < 16 * 832; ii += 32) {
        int r = ii / 832;
        int k = ii - r * 832;
        int g = tile * 16 + r;
        float v;
        if (k < 64) {
            v = input[(size_t)g * DD + k];
        } else {
            int kp = k - 64;
            int f = kp / 3;           // feature index 0..255
            int s = kp - 3 * f;       // scale index 0..2
            int d = f >> 2;
            int st = f & 3;           // 0:mean 1:max 2:min 3:std
            int n = g >> 1;
            size_t base = (size_t)g * DD + d;
            float c = cnt[n];
            float mean = sums[base] / c;
            float stat;
            if (st == 0)      stat = mean;
            else if (st == 1) stat = funkey(mxk[base]);
            else if (st == 2) stat = funkey(mnk[base]);
            else {
                float sm = sqs[base] / c;
                stat = sqrtf(fmaxf(sm - mean * mean, EPSV));
            }
            v = stat * sc[r][s];
        }
        Xs[r * XST + k] = v;
    }
    __syncthreads();

    int rsel = lane & 15;
    int hi = lane >> 4;

    v8f c0 = {}, c1 = {}, c2 = {}, c3 = {};
    const float* w0 = W_lin + (size_t)(rsel +  0) * 832;
    const float* w1 = W_lin + (size_t)(rsel + 16) * 832;
    const float* w2 = W_lin + (size_t)(rsel + 32) * 832;
    const float* w3 = W_lin + (size_t)(rsel + 48) * 832;
    const float* xrow = &Xs[rsel * XST];

    for (int k0 = 0; k0 < 832; k0 += 4) {
        int kk = k0 + 2 * hi;
        __builtin_prefetch(&w0[kk + 128], 0, 0);   // global_prefetch_b8
        v2f a  = *(const v2f*)&xrow[kk];
        v2f b0 = *(const v2f*)&w0[kk];
        v2f b1 = *(const v2f*)&w1[kk];
        v2f b2 = *(const v2f*)&w2[kk];
        v2f b3 = *(const v2f*)&w3[kk];
        c0 = __builtin_amdgcn_wmma_f32_16x16x4_f32(false, a, false, b0, (short)0, c0, false, false);
        c1 = __builtin_amdgcn_wmma_f32_16x16x4_f32(false, a, false, b1, (short)0, c1, false, false);
        c2 = __builtin_amdgcn_wmma_f32_16x16x4_f32(false, a, false, b2, (short)0, c2, false, false);
        c3 = __builtin_amdgcn_wmma_f32_16x16x4_f32(false, a, false, b3, (short)0, c3, false, false);
    }

    v8f acc[4] = {c0, c1, c2, c3};
    #pragma unroll
    for (int tI = 0; tI < 4; ++tI) {
        int col = tI * 16 + rsel;
        float bl = b_lin[col];
        #pragma unroll
        for (int v = 0; v < 8; ++v) {
            int g = tile * 16 + v + 8 * hi;
            out[(size_t)g * DD + col] = fmaxf(acc[tI][v] + bl, 0.0f);
        }
    }
}

// ---------------- host launcher ----------------
extern "C" void kernel_launch(void* const* d_in, const int* in_sizes, int n_in,
                              void* d_out, int out_size, void* d_ws, size_t ws_size,
                              hipStream_t stream) {
    const float* input       = (const float*)d_in[0];
    const float* query       = (const float*)d_in[1];
    const float* edge_weight = (const float*)d_in[2];
    const float* boundary    = (const float*)d_in[3];
    const float* W_rel       = (const float*)d_in[4];
    const float* b_rel       = (const float*)d_in[5];
    const float* W_time      = (const float*)d_in[6];
    const float* b_time      = (const float*)d_in[7];
    const float* W_lin       = (const float*)d_in[8];
    const float* b_lin       = (const float*)d_in[9];
    const int* node_in       = (const int*)d_in[10];
    const int* node_out      = (const int*)d_in[11];
    const int* relation      = (const int*)d_in[12];
    const int* edge_time     = (const int*)d_in[13];
    const int* query_time    = (const int*)d_in[14];
    float* out = (float*)d_out;

    char* ws = (char*)d_ws;
    size_t off = 0;
    auto take = [&](size_t bytes) -> void* {
        void* p = ws + off;
        off += (bytes + 255) & ~(size_t)255;
        return p;
    };
    const size_t NBD = (size_t)NN * BB * DD;
    float*    sums      = (float*)take(NBD * 4);
    float*    sqs       = (float*)take(NBD * 4);
    unsigned* mxk       = (unsigned*)take(NBD * 4);
    unsigned* mnk       = (unsigned*)take(NBD * 4);
    float*    cnt       = (float*)take(NN * 4);
    float*    degw      = (float*)take(NN * 4);
    float*    scale_raw = (float*)take(NN * 4);
    float*    scalesum  = (float*)take(256);
    float*    rel_table = (float*)take((size_t)RR * BB * DD * 4);
    float*    ttable    = (float*)take((size_t)TMAX * DD * 4);
    float*    wtab      = (float*)take((size_t)TMAX * BB * 4);

    hipLaunchKernelGGL(k_rel,  dim3((RR * BB * DD + 255) / 256), dim3(256), 0, stream,
                       query, W_rel, b_rel, rel_table);
    hipLaunchKernelGGL(k_time, dim3((TMAX * DD + 255) / 256), dim3(256), 0, stream,
                       W_time, b_time, ttable);
    hipLaunchKernelGGL(k_w,    dim3((TMAX * BB + 255) / 256), dim3(256), 0, stream,
                       query_time, wtab);
    hipLaunchKernelGGL(k_init, dim3((int)((NBD + 255) / 256)), dim3(256), 0, stream,
                       boundary, sums, sqs, mxk, mnk, cnt, degw, scalesum);
    hipLaunchKernelGGL(k_edge, dim3(EE / 8), dim3(256), 0, stream,
                       input, edge_weight, node_in, node_out, relation, edge_time,
                       rel_table, ttable, wtab, sums, sqs, mxk, mnk, cnt, degw);
    hipLaunchKernelGGL(k_scale, dim3((NN + 255) / 256), dim3(256), 0, stream,
                       degw, scale_raw, scalesum);
    hipLaunchKernelGGL(k_gemm, dim3(NN * BB / 16), dim3(32), 0, stream,
                       input, sums, sqs, mxk, mnk, cnt, scale_raw, scalesum,
                       W_lin, b_lin, out);
}